// GCNDecoder_57655640981996
// MI455X (gfx1250) — compile-verified
//
#include <hip/hip_runtime.h>
#include <hip/hip_bf16.h>

typedef float v2f __attribute__((ext_vector_type(2)));
typedef float v8f __attribute__((ext_vector_type(8)));

#define LRELU_SLOPE 0.01f

// Native non-returning fp32 atomic add at device scope (L2 RMW).
__device__ __forceinline__ void atomic_add_f32(float* p, float v) {
    asm volatile("global_atomic_add_f32 %0, %1, off scope:SCOPE_DEV"
                 :: "v"(p), "v"(v) : "memory");
}

// ---------------- degree / normalization ----------------
__global__ void k_zero_u32(unsigned* __restrict__ p, int n) {
    int i = blockIdx.x * blockDim.x + threadIdx.x;
    if (i < n) p[i] = 0u;
}

__global__ void k_count_deg(const int* __restrict__ ei, int E, unsigned* __restrict__ deg) {
    int e = blockIdx.x * blockDim.x + threadIdx.x;
    if (e < E) {
        int d = ei[E + e];  // dst row of edge_index
        atomicAdd(&deg[d], 1u);  // native u32 atomic
    }
}

__global__ void k_dinv(const unsigned* __restrict__ deg, float* __restrict__ dinv, int n) {
    int i = blockIdx.x * blockDim.x + threadIdx.x;
    if (i < n) dinv[i] = rsqrtf((float)(deg[i] + 1u));  // +1 self-loop, always > 0
}

// ---------------- fp32 WMMA GEMM: out[n_rows x NOUT] = A[n_rows x 128] * W[128 x NOUT] ----
// One block = 8 waves, covers 128 rows. Each wave: 16-row strip, all NOUT/16 col tiles.
// W is staged in LDS with K-pairs interleaved: sW[((k>>1)*NOUT + n)*2 + (k&1)]
// so a B 4x16 fragment is a single aligned ds_load_b64 per lane (WMMA-ready pair).
template <int NOUT>
__global__ __launch_bounds__(256) void k_gemm_wmma(const float* __restrict__ A,
                                                   const float* __restrict__ W,
                                                   float* __restrict__ out, int n_rows) {
    __shared__ float sW[128 * NOUT];
    const int tid = threadIdx.x;
    for (int i = tid; i < 128 * NOUT; i += 256) {
        int k = i / NOUT, n = i % NOUT;
        sW[((k >> 1) * NOUT + n) * 2 + (k & 1)] = W[i];
    }
    __syncthreads();

    const int wave = tid >> 5;
    const int lane = tid & 31;
    const int m16  = lane & 15;   // row within tile (A), col within tile (B/C)
    const int hi   = lane >> 4;   // upper half-wave flag
    const int row0 = blockIdx.x * 128 + wave * 16;
    constexpr int NT = NOUT / 16;

    v8f zero8 = {};
    v8f acc[NT];
#pragma unroll
    for (int t = 0; t < NT; ++t) acc[t] = zero8;

    // clamp row for tail block (stores are guarded; duplicated loads are harmless)
    int rA = row0 + m16;
    if (rA >= n_rows) rA = n_rows - 1;
    const float* arow = A + (size_t)rA * 128;

    for (int k0 = 0; k0 < 128; k0 += 4) {
        // A 16x4 frag: lanes 0-15 hold K=k0,k0+1 ; lanes 16-31 hold K=k0+2,k0+3
        v2f a = *(const v2f*)(arow + k0 + 2 * hi);
        const int kh = (k0 >> 1) + hi;  // K-pair index: rows (k0+2*hi, k0+2*hi+1)
#pragma unroll
        for (int t = 0; t < NT; ++t) {
            v2f b = *(const v2f*)(sW + (size_t)(kh * NOUT + t * 16 + m16) * 2);
            acc[t] = __builtin_amdgcn_wmma_f32_16x16x4_f32(
                false, a, false, b, (short)0, acc[t], false, false);
        }
    }

#pragma unroll
    for (int t = 0; t < NT; ++t) {
#pragma unroll
        for (int v = 0; v < 8; ++v) {
            int row = row0 + v + 8 * hi;  // C layout: lanes 0-15 -> M=v, lanes 16-31 -> M=v+8
            if (row < n_rows) out[(size_t)row * NOUT + t * 16 + m16] = acc[t][v];
        }
    }
}

// ---------------- self-loop init: acc[n][f] = dinv[n]^2 * h[n][f] ----------------
template <int FEAT>
__global__ void k_self_init(const float* __restrict__ h, const float* __restrict__ dinv,
                            float* __restrict__ acc, int n_rows) {
    constexpr int QPR = FEAT / 4;  // float4 quads per row
    int t = blockIdx.x * blockDim.x + threadIdx.x;
    int node = t / QPR;
    if (node >= n_rows) return;
    int fi = (t % QPR) * 4;
    float di = dinv[node];
    float w = di * di;
    float4 v = *(const float4*)(h + (size_t)node * FEAT + fi);
    float4 o = {v.x * w, v.y * w, v.z * w, v.w * w};
    *(float4*)(acc + (size_t)node * FEAT + fi) = o;
}

// ---------------- edge aggregation: acc[d] += dinv[s]*dinv[d] * h[s] ----------------
// One wave (32 lanes) per edge: edge index / norm are wave-uniform -> scalarize with
// readfirstlane so ei/dinv reads become SMEM loads; vector path does only the
// feature-row gather (one float4/float2 per lane) + native f32 atomics.
template <int FEAT>
__global__ __launch_bounds__(256) void k_edge_agg(const float* __restrict__ h,
                                                  const float* __restrict__ dinv,
                                                  const int* __restrict__ ei, int E,
                                                  float* __restrict__ acc) {
    constexpr int VW = FEAT / 32;  // floats per lane (4 for FEAT=128, 2 for FEAT=64)
    long long t = (long long)blockIdx.x * 256 + threadIdx.x;
    int e = (int)(t >> 5);
    if (e >= E) return;  // wave-uniform exit (32 | 256)
    e = __builtin_amdgcn_readfirstlane(e);
    const int lane = threadIdx.x & 31;
    int s = ei[e];
    int d = ei[E + e];
    float nrm = dinv[s] * dinv[d];
    const float* src = h + (size_t)s * FEAT + lane * VW;
    float* dst = acc + (size_t)d * FEAT + lane * VW;
    if constexpr (VW == 4) {
        float4 v = *(const float4*)src;
        atomic_add_f32(dst + 0, nrm * v.x);
        atomic_add_f32(dst + 1, nrm * v.y);
        atomic_add_f32(dst + 2, nrm * v.z);
        atomic_add_f32(dst + 3, nrm * v.w);
    } else {
        float2 v = *(const float2*)src;
        atomic_add_f32(dst + 0, nrm * v.x);
        atomic_add_f32(dst + 1, nrm * v.y);
    }
}

// ---------------- bias (+ optional LeakyReLU), in place ----------------
template <int FEAT, bool LEAKY>
__global__ void k_bias_act(float* __restrict__ acc, const float* __restrict__ b, int n_rows) {
    constexpr int QPR = FEAT / 4;
    int t = blockIdx.x * blockDim.x + threadIdx.x;
    int node = t / QPR;
    if (node >= n_rows) return;
    int fi = (t % QPR) * 4;
    float4 v  = *(const float4*)(acc + (size_t)node * FEAT + fi);
    float4 bb = *(const float4*)(b + fi);
    v.x += bb.x; v.y += bb.y; v.z += bb.z; v.w += bb.w;
    if (LEAKY) {
        v.x = v.x > 0.0f ? v.x : LRELU_SLOPE * v.x;
        v.y = v.y > 0.0f ? v.y : LRELU_SLOPE * v.y;
        v.z = v.z > 0.0f ? v.z : LRELU_SLOPE * v.z;
        v.w = v.w > 0.0f ? v.w : LRELU_SLOPE * v.w;
    }
    *(float4*)(acc + (size_t)node * FEAT + fi) = v;
}

extern "C" void kernel_launch(void* const* d_in, const int* in_sizes, int n_in,
                              void* d_out, int out_size, void* d_ws, size_t ws_size,
                              hipStream_t stream) {
    (void)n_in; (void)out_size; (void)ws_size;
    const float* z  = (const float*)d_in[0];   // [N,128]
    const int*   ei = (const int*)d_in[1];     // [2,E]
    const float* W1 = (const float*)d_in[2];   // [128,128]
    const float* b1 = (const float*)d_in[3];   // [128]
    const float* W2 = (const float*)d_in[4];   // [128,64]
    const float* b2 = (const float*)d_in[5];   // [64]

    const int N = in_sizes[0] / 128;
    const int E = in_sizes[1] / 2;

    // workspace layout
    char* ws = (char*)d_ws;
    size_t off = 0;
    auto carve = [&](size_t bytes) -> void* {
        void* p = ws + off;
        off = (off + bytes + 255) & ~(size_t)255;
        return p;
    };
    unsigned* deg  = (unsigned*)carve((size_t)N * 4);
    float*    dinv = (float*)carve((size_t)N * 4);
    float*    h1   = (float*)carve((size_t)N * 128 * 4);  // z @ W1
    float*    a1   = (float*)carve((size_t)N * 128 * 4);  // layer-1 accumulator -> h2 in place
    float*    h3   = (float*)carve((size_t)N * 64 * 4);   // h2 @ W2
    float*    outp = (float*)d_out;                        // layer-2 accumulator / final output

    const dim3 blk(256);
    const int gN = (N + 255) / 256;
    const int gE = (E + 255) / 256;
    const int gM = (N + 127) / 128;                        // GEMM row blocks
    const int gEdge = (int)(((long long)E * 32 + 255) / 256);  // wave per edge

    // degrees + normalization
    k_zero_u32 <<<gN, blk, 0, stream>>>(deg, N);
    k_count_deg<<<gE, blk, 0, stream>>>(ei, E, deg);
    k_dinv     <<<gN, blk, 0, stream>>>(deg, dinv, N);

    // ---- layer 1 ----
    k_gemm_wmma<128><<<gM, blk, 0, stream>>>(z, W1, h1, N);
    long long t128 = (long long)N * 32;  // N * FEAT/4 threads
    int gS1 = (int)((t128 + 255) / 256);
    k_self_init<128><<<gS1, blk, 0, stream>>>(h1, dinv, a1, N);
    k_edge_agg<128><<<gEdge, blk, 0, stream>>>(h1, dinv, ei, E, a1);
    k_bias_act<128, true><<<gS1, blk, 0, stream>>>(a1, b1, N);  // a1 becomes h2

    // ---- layer 2 ----
    k_gemm_wmma<64><<<gM, blk, 0, stream>>>(a1, W2, h3, N);
    long long t64 = (long long)N * 16;
    int gS2 = (int)((t64 + 255) / 256);
    k_self_init<64><<<gS2, blk, 0, stream>>>(h3, dinv, outp, N);
    k_edge_agg<64><<<gEdge, blk, 0, stream>>>(h3, dinv, ei, E, outp);
    k_bias_act<64, false><<<gS2, blk, 0, stream>>>(outp, b2, N);
}